// Encoder_33011118637330
// MI455X (gfx1250) — compile-verified
//
#include <hip/hip_runtime.h>

typedef __attribute__((ext_vector_type(2))) float v2f;
typedef __attribute__((ext_vector_type(8))) float v8f;

#define VOCAB 60
#define EMB   100
#define HID   100
#define G3    300      // 3*HID
#define CHUNK 256      // x-index staging chunk

// ---------------------------------------------------------------------------
// Kernel 1: T[v][j] = sum_e emb[v][e] * w_ih[j][e] + b_ih[j]   (fp32 WMMA)
// M=60 -> 4 tiles of 16, N=300 -> 19 tiles of 16, K=100 -> 25 steps of 4.
// One wave (32 lanes) per 16x16 output tile; EXEC is all-ones (no divergence,
// OOB handled by index clamping + guarded stores).
// A 16x4 f32 layout: lane l holds M=l&15, K-pair (l<16 ? 0,1 : 2,3).
// B 4x16 f32 layout: lane l holds N=l&15, K-pair (l<16 ? 0,1 : 2,3).
// C/D 16x16 f32: VGPR r, lane l -> row M = r + 8*(l>>4), col N = l&15.
// ---------------------------------------------------------------------------
__global__ __launch_bounds__(32) void build_table_wmma(
    const float* __restrict__ emb, const float* __restrict__ w_ih,
    const float* __restrict__ b_ih, float* __restrict__ T)
{
  const int tile = blockIdx.x;
  const int tm = tile & 3;        // 4 row tiles  (vocab)
  const int tn = tile >> 2;       // 19 col tiles (3H outputs)
  const int lane = threadIdx.x & 31;
  const int half = lane >> 4;     // K offset: 0 -> k{0,1}, 1 -> k{2,3}
  const int l16  = lane & 15;

  const int arow = tm * 16 + l16;                    // vocab row for A
  const int ar   = arow < VOCAB ? arow : VOCAB - 1;  // clamp OOB loads
  const int bcol = tn * 16 + l16;                    // output col for B/C/D
  const int bc   = bcol < G3 ? bcol : G3 - 1;

  const float* aptr = emb  + ar * EMB + half * 2;
  const float* bptr = w_ih + bc * EMB + half * 2;

  const float bias = b_ih[bc];                       // C[m][n] = b_ih[n]
  v8f c;
  #pragma unroll
  for (int r = 0; r < 8; ++r) c[r] = bias;

  #pragma unroll
  for (int kk = 0; kk < EMB; kk += 4) {
    v2f a; a.x = aptr[kk]; a.y = aptr[kk + 1];
    v2f b; b.x = bptr[kk]; b.y = bptr[kk + 1];
    // (neg_a, A, neg_b, B, c_mod, C, reuse_a, reuse_b)
    c = __builtin_amdgcn_wmma_f32_16x16x4_f32(false, a, false, b,
                                              (short)0, c, false, false);
  }

  #pragma unroll
  for (int r = 0; r < 8; ++r) {
    const int m = tm * 16 + half * 8 + r;
    if (m < VOCAB && bcol < G3) T[m * G3 + bcol] = c[r];
  }
}

// ---------------------------------------------------------------------------
// Kernel 2: the serial GRU scan. Single workgroup, 320 threads (10 wave32s).
// Thread j (<300) owns w_hh row j in 100 VGPRs; h broadcast from LDS.
// ---------------------------------------------------------------------------
__device__ __forceinline__ float fast_sigmoid(float x) {
  return 1.0f / (1.0f + __expf(-x));
}
__device__ __forceinline__ float fast_tanh(float x) {
  x = fminf(fmaxf(x, -30.0f), 30.0f);
  const float e = __expf(-2.0f * x);
  return (1.0f - e) / (1.0f + e);
}

__global__ __launch_bounds__(320, 1) void gru_scan(
    const int* __restrict__ x, const float* __restrict__ Tg,
    const float* __restrict__ w_hh, const float* __restrict__ b_hh,
    float* __restrict__ out, int seq)
{
  __shared__ float4 sT4[(VOCAB * G3) / 4];  // 72000 B: full xp table
  __shared__ float4 sh4[HID / 4];           // hidden state h (100 f32)
  __shared__ float  sm[G3];                 // per-step gate pre-activations
  __shared__ int    sx[CHUNK];              // staged indices

  const int tid = threadIdx.x;
  const int jc  = tid < G3 ? tid : G3 - 1;  // clamped row id for loads
  const bool act = tid < G3;

  // Preload this thread's w_hh row into registers (25 x float4 = 100 VGPRs).
  float4 w[25];
  const float4* wrow = (const float4*)(w_hh + jc * HID);  // 400B rows: aligned
  #pragma unroll
  for (int k = 0; k < 25; ++k) w[k] = wrow[k];
  const float bias = b_hh[jc];

  // Stage T into LDS; zero-init h.
  const float4* Tg4 = (const float4*)Tg;
  for (int i = tid; i < (VOCAB * G3) / 4; i += 320) sT4[i] = Tg4[i];
  if (tid < HID / 4) sh4[tid] = make_float4(0.f, 0.f, 0.f, 0.f);
  __syncthreads();

  const float* sT = (const float*)sT4;
  float*       sh = (float*)sh4;

  for (int base = 0; base < seq; base += CHUNK) {
    if (tid < CHUNK) sx[tid] = x[base + tid];
    __syncthreads();

    for (int s = 0; s < CHUNK; ++s) {
      const int xt = sx[s];
      const float* Trow = sT + xt * G3;     // conflict-free consecutive reads
      const float xp = Trow[jc];            // xr/xz (folded); xn handled below

      // m[j] = b_hh[j] + h . w_hh[j]   (4 accumulators break the FMA chain)
      float a0 = bias, a1 = 0.f, a2 = 0.f, a3 = 0.f;
      #pragma unroll
      for (int k = 0; k < 25; ++k) {
        const float4 hv = sh4[k];           // ds_load_b128 broadcast
        a0 = fmaf(w[k].x, hv.x, a0);
        a1 = fmaf(w[k].y, hv.y, a1);
        a2 = fmaf(w[k].z, hv.z, a2);
        a3 = fmaf(w[k].w, hv.w, a3);
      }
      float m = (a0 + a1) + (a2 + a3);
      if (tid < 2 * HID) m += xp;           // r,z gates: xp added pre-sigmoid
      if (act) sm[tid] = m;                 // n gate: keep h-part separate
      __syncthreads();

      if (tid < HID) {
        const float r  = fast_sigmoid(sm[tid]);
        const float z  = fast_sigmoid(sm[HID + tid]);
        const float n  = fast_tanh(Trow[2 * HID + tid] + r * sm[2 * HID + tid]);
        const float ho = sh[tid];
        sh[tid] = n + z * (ho - n);         // (1-z)*n + z*h
      }
      __syncthreads();
    }
  }

  if (tid < HID) out[tid] = sh[tid];
}

// ---------------------------------------------------------------------------
extern "C" void kernel_launch(void* const* d_in, const int* in_sizes, int n_in,
                              void* d_out, int out_size, void* d_ws, size_t ws_size,
                              hipStream_t stream) {
  const int*   x    = (const int*)  d_in[0];
  const float* emb  = (const float*)d_in[1];
  const float* w_ih = (const float*)d_in[2];
  const float* w_hh = (const float*)d_in[3];
  const float* b_ih = (const float*)d_in[4];
  const float* b_hh = (const float*)d_in[5];
  float* out = (float*)d_out;
  float* T   = (float*)d_ws;                    // 60*300 f32 = 72000 B scratch

  build_table_wmma<<<dim3(4 * 19), dim3(32), 0, stream>>>(emb, w_ih, b_ih, T);
  gru_scan<<<dim3(1), dim3(320), 0, stream>>>(x, T, w_hh, b_hh, out,
                                              in_sizes[0]);
}